// YoloIconCaptioner_86131274154158
// MI455X (gfx1250) — compile-verified
//
#include <hip/hip_runtime.h>

// ---------------------------------------------------------------------------
// Fused YOLO post-process for MI455X (gfx1250, wave32).
// Single workgroup (512 thr = 16 waves) — the problem is latency bound:
// ~0.7MB total traffic (~0.03us @ 23.3TB/s), so we fuse decode+NMS+ROI into
// one kernel, keep all 8400 anchors register-resident, prefetch the 0.4MB
// feature map into L2 (global_prefetch_b8) while the serial 10-round NMS
// runs, and execute ROI-align as a per-box 16x36x256 f32 GEMM on the WMMA
// pipe (V_WMMA_F32_16X16X4_F32, K=36 taps -> 9 chained k-steps).
// ---------------------------------------------------------------------------

#define IMGSZ    640
#define NA       8400
#define FH       20          // feature H == W
#define FC       256         // channels
#define NTOPK    10
#define CONF_TH  0.25f
#define IOU_TH   0.45f
#define NTHR     512
#define NWAVE    (NTHR / 32)
#define SLOTS    17          // ceil(8400/512)
#define VSTRIDE  272         // 272 % 64 == 16 -> lane-halves hit disjoint LDS banks

typedef __attribute__((ext_vector_type(2))) float v2f;
typedef __attribute__((ext_vector_type(8))) float v8f;

static __device__ __forceinline__ int iclamp(int v, int lo, int hi) {
  return v < lo ? lo : (v > hi ? hi : v);
}

__global__ void __launch_bounds__(NTHR, 1)
yolo_fused_kernel(const float* __restrict__ pred,
                  const float* __restrict__ feat,
                  float* __restrict__ out)
{
  __shared__ float s_rv[NWAVE];
  __shared__ int   s_ri[NWAVE];
  __shared__ float s_win_box[4];
  __shared__ int   s_win_idx;
  __shared__ float s_win_val;
  __shared__ float s_keep[NTOPK][4];
  __shared__ float s_ok[NTOPK];
  __shared__ int   s_tap_pos[36];
  __shared__ float s_tap_w[36];
  __shared__ float sV[36 * VSTRIDE];   // 36 tap rows x 256 ch (padded), ~39KB

  const int tid  = threadIdx.x;
  const int wave = tid >> 5;
  const int lane = tid & 31;
  const float NEG = -__builtin_inff();

  // ---- Phase 0: decode anchors into registers (coalesced) -----------------
  float bx1[SLOTS], by1[SLOTS], bx2[SLOTS], by2[SLOTS], sc[SLOTS];
  #pragma unroll
  for (int j = 0; j < SLOTS; ++j) {
    int a = tid + j * NTHR;
    if (a < NA) {
      float cx = pred[a];
      float cy = pred[NA + a];
      float w  = pred[2 * NA + a];
      float h  = pred[3 * NA + a];
      float cf = pred[4 * NA + a];
      bx1[j] = cx - 0.5f * w;  by1[j] = cy - 0.5f * h;
      bx2[j] = cx + 0.5f * w;  by2[j] = cy + 0.5f * h;
      sc[j]  = (cf >= CONF_TH) ? cf : NEG;     // valid0 mask folded into score
    } else {
      bx1[j] = by1[j] = bx2[j] = by2[j] = 0.f;
      sc[j] = NEG;
    }
  }

  // Warm L2 with the whole feature map while NMS serializes.
  // Emits gfx1250 global_prefetch_b8 (speculative, no counters).
  for (int i = tid; i < (FC * FH * FH * 4) / 128; i += NTHR)
    __builtin_prefetch(((const char*)feat) + i * 128, 0, 1);

  // ---- Phase 1: greedy NMS, NTOPK serial rounds ---------------------------
  for (int k = 0; k < NTOPK; ++k) {
    // local argmax (lowest-index tie-break, matches jnp.argmax)
    float bv = sc[0]; int bi = tid;
    float lx1 = bx1[0], ly1 = by1[0], lx2 = bx2[0], ly2 = by2[0];
    #pragma unroll
    for (int j = 1; j < SLOTS; ++j) {
      if (sc[j] > bv) {
        bv = sc[j]; bi = tid + j * NTHR;
        lx1 = bx1[j]; ly1 = by1[j]; lx2 = bx2[j]; ly2 = by2[j];
      }
    }
    // wave32 butterfly argmax
    #pragma unroll
    for (int off = 16; off; off >>= 1) {
      float ov = __shfl_xor(bv, off, 32);
      int   oi = __shfl_xor(bi, off, 32);
      if (ov > bv || (ov == bv && oi < bi)) { bv = ov; bi = oi; }
    }
    if (lane == 0) { s_rv[wave] = bv; s_ri[wave] = bi; }
    __syncthreads();
    if (wave == 0) {
      float v = (lane < NWAVE) ? s_rv[lane] : NEG;
      int   i = (lane < NWAVE) ? s_ri[lane] : 0x7fffffff;
      #pragma unroll
      for (int off = 16; off; off >>= 1) {
        float ov = __shfl_xor(v, off, 32);
        int   oi = __shfl_xor(i, off, 32);
        if (ov > v || (ov == v && oi < i)) { v = ov; i = oi; }
      }
      if (lane == 0) { s_win_idx = i; s_win_val = v; }
    }
    __syncthreads();
    const int  widx = s_win_idx;
    const bool has  = s_win_val > NEG;          // masked[i] > -inf
    if (tid == (widx & (NTHR - 1))) {           // owner's local best IS the winner
      s_win_box[0] = lx1; s_win_box[1] = ly1;
      s_win_box[2] = lx2; s_win_box[3] = ly2;
    }
    __syncthreads();
    const float wx1 = s_win_box[0], wy1 = s_win_box[1];
    const float wx2 = s_win_box[2], wy2 = s_win_box[3];
    if (tid == 0) {
      float okf = has ? 1.f : 0.f;
      s_ok[k] = okf;
      s_keep[k][0] = wx1 * okf; s_keep[k][1] = wy1 * okf;
      s_keep[k][2] = wx2 * okf; s_keep[k][3] = wy2 * okf;
    }
    if (has) {
      float wa = fmaxf(wx2 - wx1, 0.f) * fmaxf(wy2 - wy1, 0.f);
      #pragma unroll
      for (int j = 0; j < SLOTS; ++j) {
        int a = tid + j * NTHR;
        float ltx = fmaxf(bx1[j], wx1), lty = fmaxf(by1[j], wy1);
        float rbx = fminf(bx2[j], wx2), rby = fminf(by2[j], wy2);
        float inter = fmaxf(rbx - ltx, 0.f) * fmaxf(rby - lty, 0.f);
        float aj  = fmaxf(bx2[j] - bx1[j], 0.f) * fmaxf(by2[j] - by1[j], 0.f);
        float uni = fmaxf(aj + wa - inter, 1e-9f);
        if (inter / uni > IOU_TH || a == widx) sc[j] = NEG;
      }
    }
  }
  __syncthreads();

  // ---- Phase 2: ROI-align as per-box WMMA GEMM ----------------------------
  // tokens[b] (9x256) = W (9x36 bilinear weights) x V (36 tap rows x 256 ch)
  const float scale = (float)FH / (float)IMGSZ;
  for (int b = 0; b < NTOPK; ++b) {
    if (tid < 36) {                              // tap table: pos + weight
      int kk = tid, s = kk >> 2, tap = kk & 3;
      float x1 = s_keep[b][0] * scale, y1 = s_keep[b][1] * scale;
      float x2 = s_keep[b][2] * scale, y2 = s_keep[b][3] * scale;
      float bw = fmaxf(x2 - x1, 1e-6f) * (1.f / 3.f);
      float bh = fmaxf(y2 - y1, 1e-6f) * (1.f / 3.f);
      int gy = s / 3, gx = s - gy * 3;
      float xs = x1 + ((float)gx + 0.5f) * bw;
      float ys = y1 + ((float)gy + 0.5f) * bh;
      float fx = floorf(xs), fy = floorf(ys);
      float wx = xs - fx, wy = ys - fy;
      int x0 = iclamp((int)fx, 0, FH - 1), y0 = iclamp((int)fy, 0, FH - 1);
      int x1i = iclamp(x0 + 1, 0, FH - 1), y1i = iclamp(y0 + 1, 0, FH - 1);
      int ta = tap >> 1, tb = tap & 1;
      s_tap_pos[kk] = (ta ? y1i : y0) * FH + (tb ? x1i : x0);
      s_tap_w[kk]   = (ta ? wy : 1.f - wy) * (tb ? wx : 1.f - wx);
    }
    __syncthreads();
    for (int idx = tid; idx < 36 * FC; idx += NTHR) {   // gather V into LDS
      int kk = idx >> 8, ch = idx & 255;
      sV[kk * VSTRIDE + ch] = feat[ch * (FH * FH) + s_tap_pos[kk]];
    }
    __syncthreads();
    {
      const int ntile = wave;            // 16 waves x 16 channels = 256
      const int row = lane & 15, hi = lane >> 4;
      v8f acc = {0.f, 0.f, 0.f, 0.f, 0.f, 0.f, 0.f, 0.f};
      #pragma unroll
      for (int kk = 0; kk < 9; ++kk) {   // K = 36 = 9 x (K=4) WMMA steps
        // A 16x4 layout: lanes 0-15 K={0,1}, lanes 16-31 K={2,3}; row = M
        v2f av, bvv;
        float w0 = s_tap_w[4 * kk + (hi ? 2 : 0)];
        float w1 = s_tap_w[4 * kk + (hi ? 3 : 1)];
        av.x = (row == kk) ? w0 : 0.f;   // W is block-diagonal: sample kk owns taps 4kk..4kk+3
        av.y = (row == kk) ? w1 : 0.f;
        // B 4x16: VGPR0 lanes 0-15 K=0 / 16-31 K=1, VGPR1 K=2/K=3
        bvv.x = sV[(4 * kk + (hi ? 1 : 0)) * VSTRIDE + ntile * 16 + row];
        bvv.y = sV[(4 * kk + (hi ? 3 : 2)) * VSTRIDE + ntile * 16 + row];
        acc = __builtin_amdgcn_wmma_f32_16x16x4_f32(
                  /*neg_a=*/false, av, /*neg_b=*/false, bvv,
                  /*c_mod=*/(short)0, acc, /*reuse_a=*/false, /*reuse_b=*/false);
      }
      const float okf = s_ok[b];
      #pragma unroll
      for (int v = 0; v < 8; ++v) {      // D: VGPR v -> rows v / v+8
        int M = v + (hi ? 8 : 0);
        if (M < 9)
          out[(b * 9 + M) * FC + ntile * 16 + row] = acc[v] * okf;
      }
    }
    __syncthreads();
  }
}

extern "C" void kernel_launch(void* const* d_in, const int* in_sizes, int n_in,
                              void* d_out, int out_size, void* d_ws, size_t ws_size,
                              hipStream_t stream) {
  (void)in_sizes; (void)n_in; (void)d_ws; (void)ws_size; (void)out_size;
  const float* pred = (const float*)d_in[0];   // (1,5,8400) f32
  const float* feat = (const float*)d_in[1];   // (1,256,20,20) f32
  float* out = (float*)d_out;                  // (10,9,256) f32
  hipLaunchKernelGGL(yolo_fused_kernel, dim3(1), dim3(NTHR), 0, stream,
                     pred, feat, out);
}